// SelectDillDeepFuse_22024592294390
// MI455X (gfx1250) — compile-verified
//
#include <hip/hip_runtime.h>
#include <hip/hip_bf16.h>

typedef float v2f __attribute__((ext_vector_type(2)));
typedef float v8f __attribute__((ext_vector_type(8)));

#define NTOK  1024
#define CEMB  384
#define WROWS 32
#define BATCH 32

// ---------------------------------------------------------------------------
// Pack px(8) | prx(8) | pm(8) | prm(8) -> Wall[32][384]; gx,gm -> Gwall[2][384]
// ---------------------------------------------------------------------------
__global__ __launch_bounds__(384) void prep_wall(
    const float* __restrict__ gx_w, const float* __restrict__ gm_w,
    const float* __restrict__ px_w, const float* __restrict__ prx_w,
    const float* __restrict__ pm_w, const float* __restrict__ prm_w,
    float* __restrict__ wall, float* __restrict__ gwall) {
  int r = blockIdx.x;    // 0..33
  int c = threadIdx.x;   // 0..383
  if      (r < 8)  wall[r * CEMB + c] = px_w[r * CEMB + c];
  else if (r < 16) wall[r * CEMB + c] = prx_w[(r - 8) * CEMB + c];
  else if (r < 24) wall[r * CEMB + c] = pm_w[(r - 16) * CEMB + c];
  else if (r < 32) wall[r * CEMB + c] = prm_w[(r - 24) * CEMB + c];
  else if (r == 32) gwall[c] = gx_w[c];
  else              gwall[CEMB + c] = gm_w[c];
}

// ---------------------------------------------------------------------------
// Proj[b][half] = Wall(32x384) @ x[b,half](384x1024), fp32 WMMA 16x16x4.
// One wave per (b, half, 16-token tile): 2 row-tiles (no padding waste),
// plus the branch score row (gx for half 0, gm for half 1) accumulated with
// v_fmac from the same in-register B values, reduced across lane halves.
// A layout : lane m=lane&15, lanes0-15 hold K={k0,k0+1}, lanes16-31 K={k0+2,k0+3}
// B layout : lane n=lane&15, same K-half split (rows striped across lanes)
// D layout : vgpr v -> M=v (lanes0-15) / M=v+8 (lanes16-31), N=lane&15
// ---------------------------------------------------------------------------
__global__ __launch_bounds__(128) void gemm_wmma(
    const float* __restrict__ x, const float* __restrict__ wall,
    const float* __restrict__ gwall,
    float* __restrict__ proj, float* __restrict__ scores) {
  __shared__ float sW[WROWS * CEMB];   // 49.2 KB
  __shared__ float sG[2 * CEMB];       // 3 KB
  for (int i = threadIdx.x; i < WROWS * CEMB; i += 128) sW[i] = wall[i];
  for (int i = threadIdx.x; i < 2 * CEMB; i += 128)     sG[i] = gwall[i];
  __syncthreads();

  int wave = blockIdx.x * 4 + (threadIdx.x >> 5);
  int lane = threadIdx.x & 31;
  int mat  = wave & 1;          // 0 = xt half, 1 = mt half
  int nt   = (wave >> 1) & 63;  // token tile
  int b    = wave >> 7;         // batch

  const float* Bb = x    + (size_t)(b * 2 + mat) * CEMB * NTOK + nt * 16;
  float*       Ob = proj + (size_t)(b * 2 + mat) * WROWS * NTOK + nt * 16;

  int lh = lane >> 4;   // lane half selects K pair
  int ll = lane & 15;   // m for A, n for B/D

  const float* gw = sG + mat * CEMB;

  v8f c0 = {}, c1 = {};
  float sacc = 0.0f;
  for (int k0 = 0; k0 < CEMB; k0 += 4) {
    int kb = k0 + lh * 2;
    v2f bm;
    bm.x = Bb[(size_t)kb * NTOK + ll];
    bm.y = Bb[(size_t)(kb + 1) * NTOK + ll];
    const float* wa = sW + ll * CEMB + kb;      // 8B aligned (kb even)
    v2f a0 = *(const v2f*)(wa);
    v2f a1 = *(const v2f*)(wa + 16 * CEMB);
    v2f gv = *(const v2f*)(gw + kb);
    c0 = __builtin_amdgcn_wmma_f32_16x16x4_f32(false, a0, false, bm, (short)0, c0, false, false);
    c1 = __builtin_amdgcn_wmma_f32_16x16x4_f32(false, a1, false, bm, (short)0, c1, false, false);
    sacc = fmaf(bm.x, gv.x, sacc);
    sacc = fmaf(bm.y, gv.y, sacc);
  }
#pragma unroll
  for (int v = 0; v < 8; v++) {
    Ob[(size_t)(v + lh * 8) * NTOK + ll]      = c0[v];
    Ob[(size_t)(16 + v + lh * 8) * NTOK + ll] = c1[v];
  }
  // lane ll and lane ll+16 hold disjoint K halves of token n = nt*16 + ll
  float tot = sacc + __shfl_xor(sacc, 16, 32);
  if (lh == 0) {
    scores[((size_t)mat * BATCH + b) * NTOK + nt * 16 + ll] = tot;
  }
}

// ---------------------------------------------------------------------------
// Stable top-256 / bottom-256 masks for both branches (softmax is monotonic
// => rank raw scores). One block per batch, O(N^2) rank counting in LDS.
// ---------------------------------------------------------------------------
__global__ __launch_bounds__(1024) void mask_kernel(
    const float* __restrict__ scores, const float* __restrict__ gx_b,
    const float* __restrict__ gm_b, float* __restrict__ masks) {
  __shared__ float sx[NTOK];
  __shared__ float sm[NTOK];
  int b = blockIdx.x, n = threadIdx.x;
  float s_x = scores[((size_t)0 * BATCH + b) * NTOK + n] + gx_b[0];
  float s_m = scores[((size_t)1 * BATCH + b) * NTOK + n] + gm_b[0];
  sx[n] = s_x; sm[n] = s_m;
  __syncthreads();
  int gtx = 0, ltx = 0, gtm = 0, ltm = 0, eqx = 0, eqm = 0;
  for (int j = 0; j < NTOK; j++) {
    float vx = sx[j], vm = sm[j];
    gtx += (vx > s_x); ltx += (vx < s_x);
    gtm += (vm > s_m); ltm += (vm < s_m);
    if (j < n) { eqx += (vx == s_x); eqm += (vm == s_m); }
  }
  float4 mk;
  mk.x = ((gtx + eqx) < 256) ? 1.0f : 0.0f;   // tm_x
  mk.y = ((ltx + eqx) < 256) ? 1.0f : 0.0f;   // rm_x
  mk.z = ((gtm + eqm) < 256) ? 1.0f : 0.0f;   // tm_m
  mk.w = ((ltm + eqm) < 256) ? 1.0f : 0.0f;   // rm_m
  ((float4*)masks)[(size_t)b * NTOK + n] = mk;
}

// ---------------------------------------------------------------------------
// Per-token scalar recombination + 16->8 fuse + 8->384 expansion.
// Wall row map: 0..7 px | 8..15 prx | 16..23 pm | 24..31 prm
// ---------------------------------------------------------------------------
__global__ __launch_bounds__(256) void combine_kernel(
    const float* __restrict__ proj, const float* __restrict__ masks,
    const float* __restrict__ px_b, const float* __restrict__ prx_b,
    const float* __restrict__ pm_b, const float* __restrict__ prm_b,
    const float* __restrict__ pfx_w, const float* __restrict__ pfx_b,
    const float* __restrict__ pfm_w, const float* __restrict__ pfm_b,
    const float* __restrict__ pb_w, const float* __restrict__ pb_b,
    float* __restrict__ out) {
  __shared__ float s_pbw[CEMB * 8];
  __shared__ float s_pbb[CEMB];
  __shared__ float s_pfx[128];
  __shared__ float s_pfm[128];
  __shared__ float s_lb[32];     // px_b | prx_b | pm_b | prm_b
  __shared__ float s_fbx[8];
  __shared__ float s_fbm[8];
  int t = threadIdx.x;
  for (int i = t; i < CEMB * 8; i += 256) s_pbw[i] = pb_w[i];
  for (int i = t; i < CEMB; i += 256)     s_pbb[i] = pb_b[i];
  if (t < 128) { s_pfx[t] = pfx_w[t]; s_pfm[t] = pfm_w[t]; }
  if (t < 8) {
    s_lb[t] = px_b[t]; s_lb[8 + t] = prx_b[t];
    s_lb[16 + t] = pm_b[t]; s_lb[24 + t] = prm_b[t];
    s_fbx[t] = pfx_b[t]; s_fbm[t] = pfm_b[t];
  }
  __syncthreads();

  int gid = blockIdx.x * 256 + t;
  int b = gid >> 10, n = gid & 1023;
  const float* PX = proj + (size_t)(b * 2 + 0) * WROWS * NTOK + n;
  const float* PM = proj + (size_t)(b * 2 + 1) * WROWS * NTOK + n;
  float4 mk = ((const float4*)masks)[gid];
  float midx = 1.0f - mk.x - mk.y;
  float midm = 1.0f - mk.z - mk.w;

  float lx[16], lm[16];
#pragma unroll
  for (int i = 0; i < 8; i++) {
    // low_x = tm_x*(xt@px) + rm_x*(mt@px) + px_b
    lx[i] = mk.x * PX[(size_t)(0 + i) * NTOK] + mk.y * PM[(size_t)(0 + i) * NTOK] + s_lb[i];
    // low_res_x = mid_x*((xt+mt)@prx) + prx_b
    lx[8 + i] = midx * (PX[(size_t)(8 + i) * NTOK] + PM[(size_t)(8 + i) * NTOK]) + s_lb[8 + i];
    // low_m = tm_m*(mt@pm) + rm_m*(xt@pm) + pm_b
    lm[i] = mk.z * PM[(size_t)(16 + i) * NTOK] + mk.w * PX[(size_t)(16 + i) * NTOK] + s_lb[16 + i];
    // low_res_m = mid_m*((xt+mt)@prm) + prm_b
    lm[8 + i] = midm * (PX[(size_t)(24 + i) * NTOK] + PM[(size_t)(24 + i) * NTOK]) + s_lb[24 + i];
  }

  float g[8];
#pragma unroll
  for (int o = 0; o < 8; o++) {
    float ax = s_fbx[o], am = s_fbm[o];
#pragma unroll
    for (int i = 0; i < 16; i++) {
      ax += s_pfx[o * 16 + i] * lx[i];
      am += s_pfm[o * 16 + i] * lm[i];
    }
    g[o] = ax + am;
  }

  float* ob = out + (size_t)b * CEMB * NTOK + n;
  for (int e = 0; e < CEMB; e++) {
    float acc = s_pbb[e];
#pragma unroll
    for (int r = 0; r < 8; r++) acc += s_pbw[e * 8 + r] * g[r];
    ob[(size_t)e * NTOK] = acc;
  }
}

// ---------------------------------------------------------------------------
extern "C" void kernel_launch(void* const* d_in, const int* in_sizes, int n_in,
                              void* d_out, int out_size, void* d_ws, size_t ws_size,
                              hipStream_t stream) {
  const float* x     = (const float*)d_in[0];
  const float* gx_w  = (const float*)d_in[1];
  const float* gx_b  = (const float*)d_in[2];
  const float* gm_w  = (const float*)d_in[3];
  const float* gm_b  = (const float*)d_in[4];
  const float* px_w  = (const float*)d_in[5];
  const float* px_b  = (const float*)d_in[6];
  const float* pm_w  = (const float*)d_in[7];
  const float* pm_b  = (const float*)d_in[8];
  const float* prx_w = (const float*)d_in[9];
  const float* prx_b = (const float*)d_in[10];
  const float* prm_w = (const float*)d_in[11];
  const float* prm_b = (const float*)d_in[12];
  const float* pfx_w = (const float*)d_in[13];
  const float* pfx_b = (const float*)d_in[14];
  const float* pfm_w = (const float*)d_in[15];
  const float* pfm_b = (const float*)d_in[16];
  const float* pb_w  = (const float*)d_in[17];
  const float* pb_b  = (const float*)d_in[18];
  float* out = (float*)d_out;

  float* ws     = (float*)d_ws;
  float* wall   = ws;                                         // 32*384      = 12,288 f
  float* gwall  = wall + WROWS * CEMB;                        // 2*384       = 768 f
  float* proj   = gwall + 2 * CEMB;                           // 32*2*32*1024= 2,097,152 f
  float* scores = proj + (size_t)BATCH * 2 * WROWS * NTOK;    // 2*32*1024   = 65,536 f
  float* masks  = scores + (size_t)2 * BATCH * NTOK;          // 32*1024*4   = 131,072 f

  prep_wall<<<34, CEMB, 0, stream>>>(gx_w, gm_w, px_w, prx_w, pm_w, prm_w, wall, gwall);
  // 32 batches * 2 halves * 64 token-tiles = 4096 waves, 4 waves/block
  gemm_wmma<<<1024, 128, 0, stream>>>(x, wall, gwall, proj, scores);
  mask_kernel<<<BATCH, NTOK, 0, stream>>>(scores, gx_b, gm_b, masks);
  combine_kernel<<<(BATCH * NTOK) / 256, 256, 0, stream>>>(
      proj, masks, px_b, prx_b, pm_b, prm_b,
      pfx_w, pfx_b, pfm_w, pfm_b, pb_w, pb_b, out);
}